// Net_87325275062555
// MI455X (gfx1250) — compile-verified
//
#include <hip/hip_runtime.h>

typedef __attribute__((ext_vector_type(2))) float v2f;
typedef __attribute__((ext_vector_type(8))) float v8f;

#define BETA    0.9f
#define THRESH  1.0f
#define NSTEPS  50
#define NB      2048
#define NIN     1024
#define NHID    512
#define NOUT    10
#define NOUTP   16
#define M2      (NSTEPS * NB)     // 102400 rows for batched layer-2 GEMM

// -------------------------------------------------------------------------
// GEMM1: cur1[m,n] = sum_k x[m,k] * W1[n,k]     (M=2048, K=1024, N=512)
// One wave -> 16x64 output tile (4 accumulators, A-fragment reused 4x).
// 128 blocks * 8 waves = 1024 waves = 128 Mtiles * 8 Ngroups.
// -------------------------------------------------------------------------
__global__ __launch_bounds__(256) void gemm1_wmma(
    const float* __restrict__ x, const float* __restrict__ W1,
    float* __restrict__ cur1) {
  const int lane  = threadIdx.x & 31;
  const int wave  = threadIdx.x >> 5;
  const int gw    = blockIdx.x * 8 + wave;
  const int mtile = gw >> 3;                 // 0..127
  const int ngrp  = gw & 7;                  // 0..7 (64 cols each)
  const int m     = (mtile << 4) + (lane & 15);
  const int klane = (lane >> 4) << 1;        // 0 or 2
  const int ncol  = (ngrp << 6) + (lane & 15);

  v8f a0 = {}, a1 = {}, a2 = {}, a3 = {};
  const float* xrow = x + (size_t)m * NIN;
  const float* w0 = W1 + (size_t)(ncol +  0) * NIN;
  const float* w1 = W1 + (size_t)(ncol + 16) * NIN;
  const float* w2 = W1 + (size_t)(ncol + 32) * NIN;
  const float* w3 = W1 + (size_t)(ncol + 48) * NIN;

  for (int k = 0; k < NIN; k += 4) {
    v2f af = *(const v2f*)(xrow + k + klane);
    v2f b0 = *(const v2f*)(w0 + k + klane);
    v2f b1 = *(const v2f*)(w1 + k + klane);
    v2f b2 = *(const v2f*)(w2 + k + klane);
    v2f b3 = *(const v2f*)(w3 + k + klane);
    a0 = __builtin_amdgcn_wmma_f32_16x16x4_f32(false, af, false, b0, (short)0, a0, false, false);
    a1 = __builtin_amdgcn_wmma_f32_16x16x4_f32(false, af, false, b1, (short)0, a1, false, false);
    a2 = __builtin_amdgcn_wmma_f32_16x16x4_f32(false, af, false, b2, (short)0, a2, false, false);
    a3 = __builtin_amdgcn_wmma_f32_16x16x4_f32(false, af, false, b3, (short)0, a3, false, false);
  }

  const int half = lane >> 4;
  const int col  = lane & 15;
  #pragma unroll
  for (int r = 0; r < 8; ++r) {
    const size_t row = (size_t)((mtile << 4) + r + half * 8) * NHID + (ngrp << 6) + col;
    cur1[row +  0] = a0[r];
    cur1[row + 16] = a1[r];
    cur1[row + 32] = a2[r];
    cur1[row + 48] = a3[r];
  }
}

// -------------------------------------------------------------------------
// Layer-1 LIF dynamics: elementwise per (b,h); membrane kept in a register
// across all 50 steps; per-step writes are coalesced within each t-plane.
// spk1_rec stays RT (re-read by gemm2, want L2 residency);
// mem1_rec is write-once dead data -> non-temporal to spare L2.
// -------------------------------------------------------------------------
__global__ __launch_bounds__(256) void sim_layer1(
    const float* __restrict__ cur1ws, const float* __restrict__ b1,
    float* __restrict__ spk1_rec, float* __restrict__ mem1_rec) {
  const int idx = blockIdx.x * 256 + threadIdx.x;   // < 2048*512
  const int h   = idx & (NHID - 1);
  const float cur = cur1ws[idx] + b1[h];
  float mem = 0.0f;
  #pragma unroll 5
  for (int t = 0; t < NSTEPS; ++t) {
    const float rst = (mem > THRESH) ? THRESH : 0.0f;  // spike of prev mem, detached
    mem = BETA * mem + cur - rst;
    const float spk = (mem > THRESH) ? 1.0f : 0.0f;
    const size_t o = (size_t)t * (NB * NHID) + idx;
    spk1_rec[o] = spk;                                 // RT: feeds gemm2 via L2
    __builtin_nontemporal_store(mem, mem1_rec + o);    // NT: never re-read
  }
}

// -------------------------------------------------------------------------
// GEMM2: cur2p[m,n] = sum_k spk1[m,k] * W2[n,k]   (M=102400, K=512, N=16pad)
// W2 zero-padded 10->16 rows in LDS (uniform loads, full EXEC at WMMA).
// One wave -> 32 rows x 16 cols (B fragment reused 2x).
// 400 blocks * 8 waves = 3200 waves * 32 rows = 102400.
// spk1 reads are last-use -> non-temporal loads (hit L2, then let go).
// -------------------------------------------------------------------------
__global__ __launch_bounds__(256) void gemm2_wmma(
    const float* __restrict__ spk1, const float* __restrict__ W2,
    float* __restrict__ cur2p) {
  __shared__ float w2p[NOUTP * NHID];        // 32 KB
  for (int j = threadIdx.x; j < NOUTP * NHID; j += 256) {
    const int n = j >> 9, k = j & (NHID - 1);
    w2p[j] = (n < NOUT) ? W2[n * NHID + k] : 0.0f;
  }
  __syncthreads();

  const int lane  = threadIdx.x & 31;
  const int wave  = threadIdx.x >> 5;
  const int gw    = blockIdx.x * 8 + wave;
  const int mbase = gw << 5;                 // 32 rows per wave
  const int mrow  = lane & 15;
  const int klane = (lane >> 4) << 1;
  const int n     = lane & 15;

  v8f a0 = {}, a1 = {};
  const float* s0 = spk1 + (size_t)(mbase +  0 + mrow) * NHID;
  const float* s1 = spk1 + (size_t)(mbase + 16 + mrow) * NHID;
  const float* wl = w2p + n * NHID;

  for (int k = 0; k < NHID; k += 4) {
    v2f bf = *(const v2f*)(wl + k + klane);  // LDS (ds_load_b64)
    v2f f0 = __builtin_nontemporal_load((const v2f*)(s0 + k + klane));
    v2f f1 = __builtin_nontemporal_load((const v2f*)(s1 + k + klane));
    a0 = __builtin_amdgcn_wmma_f32_16x16x4_f32(false, f0, false, bf, (short)0, a0, false, false);
    a1 = __builtin_amdgcn_wmma_f32_16x16x4_f32(false, f1, false, bf, (short)0, a1, false, false);
  }

  const int half = lane >> 4;
  const int col  = lane & 15;
  #pragma unroll
  for (int r = 0; r < 8; ++r) {
    const int rr = r + half * 8;
    cur2p[(size_t)(mbase +  0 + rr) * NOUTP + col] = a0[r];
    cur2p[(size_t)(mbase + 16 + rr) * NOUTP + col] = a1[r];
  }
}

// -------------------------------------------------------------------------
// Layer-2 LIF recurrence: 2048*10 lanes, sequential over t in registers.
// -------------------------------------------------------------------------
__global__ __launch_bounds__(256) void sim_layer2(
    const float* __restrict__ cur2p, const float* __restrict__ b2,
    float* __restrict__ spk2_rec, float* __restrict__ mem2_rec) {
  const int idx = blockIdx.x * 256 + threadIdx.x;   // < 2048*10
  const int b   = idx / NOUT;
  const int n   = idx - b * NOUT;
  const float bias = b2[n];
  float mem = 0.0f;
  #pragma unroll 5
  for (int t = 0; t < NSTEPS; ++t) {
    const float cur = cur2p[((size_t)t * NB + b) * NOUTP + n] + bias;
    const float rst = (mem > THRESH) ? THRESH : 0.0f;
    mem = BETA * mem + cur - rst;
    const float spk = (mem > THRESH) ? 1.0f : 0.0f;
    const size_t o = (size_t)t * (NB * NOUT) + idx;
    spk2_rec[o] = spk;
    mem2_rec[o] = mem;
  }
}

extern "C" void kernel_launch(void* const* d_in, const int* in_sizes, int n_in,
                              void* d_out, int out_size, void* d_ws, size_t ws_size,
                              hipStream_t stream) {
  (void)in_sizes; (void)n_in; (void)out_size; (void)ws_size;
  const float* x  = (const float*)d_in[0];   // [2048,1024]
  const float* W1 = (const float*)d_in[1];   // [512,1024]
  const float* b1 = (const float*)d_in[2];   // [512]
  const float* W2 = (const float*)d_in[3];   // [10,512]
  const float* b2 = (const float*)d_in[4];   // [10]

  float* out = (float*)d_out;
  // return order: spk2_rec, mem2_rec, spk1_rec, mem1_rec
  float* spk2 = out;                                   // 50*2048*10
  float* mem2 = out + (size_t)NSTEPS * NB * NOUT;      // +1,024,000
  float* spk1 = mem2 + (size_t)NSTEPS * NB * NOUT;     // 50*2048*512
  float* mem1 = spk1 + (size_t)NSTEPS * NB * NHID;

  float* cur1  = (float*)d_ws;                                  // 4 MB
  float* cur2p = (float*)((char*)d_ws + (size_t)NB * NHID * 4); // 6.4 MB

  gemm1_wmma<<<128, 256, 0, stream>>>(x, W1, cur1);
  sim_layer1<<<(NB * NHID) / 256, 256, 0, stream>>>(cur1, b1, spk1, mem1);
  gemm2_wmma<<<M2 / (8 * 32), 256, 0, stream>>>(spk1, W2, cur2p);
  sim_layer2<<<(NB * NOUT) / 256, 256, 0, stream>>>(cur2p, b2, spk2, mem2);
}